// LinkPrediction_6047313953619
// MI455X (gfx1250) — compile-verified
//
#include <hip/hip_runtime.h>
#include <hip/hip_bf16.h>

typedef __attribute__((ext_vector_type(2))) float v2f;
typedef __attribute__((ext_vector_type(8))) float v8f;

#define HIDDEN   128
#define K_IN     256      // 2*HIDDEN  (rows of W1 / concat length)
#define H_MID    512      // 4*HIDDEN  (cols of W1 / rows of W2)
#define N_EDGES  500000
#define N_TILES  (N_EDGES / 16)   // 31250, exact

// LDS column strides (dwords), padded so the 4 columns hit banks 0/8/16/24
#define W2_STRIDE   520   // 520 % 64 == 8
#define WEFF_STRIDE 264   // 264 % 64 == 8

// Workspace layout (floats):
//  ws[0 .. 1023]    : W_effCM column-major padded to 4 cols:
//                     ws[c*256 + k] = (c<2) ? (W1@W2)[k][c] : 0
//  ws[1024 .. 1025] : b_eff[c] = b1@W2[:,c] + b2[c]

// ---------------------------------------------------------------------------
// Kernel 1: fold W_eff = W1 (256x512) @ W2 (512x2 padded to 16 cols) with
// v_wmma_f32_16x16x4_f32.  One block, 16 waves, one 16-row M-tile per wave.
// W2 is staged zero-padded in LDS so B-fragment loads are branchless ds_load.
// ---------------------------------------------------------------------------
__global__ __launch_bounds__(512) void fold_weights_kernel(
    const float* __restrict__ W1, const float* __restrict__ bias1,
    const float* __restrict__ W2, const float* __restrict__ bias2,
    float* __restrict__ ws)
{
    __shared__ float w2cm[4 * W2_STRIDE];   // [c][k], cols 2..3 = 0

    const int tid  = threadIdx.x;
    const int wave = tid >> 5;
    const int lane = tid & 31;
    const int lrow = lane & 15;        // M (for A) / N (for B)
    const int hi   = lane >> 4;        // lane-group: K offset +2
    const int koff = hi * 2;
    const int m    = wave * 16 + lrow; // A row (same for both lane groups)

    // Stage zero-padded W2 column-major: 4*512 entries, 512 threads x 4
#pragma unroll
    for (int i = 0; i < 4; ++i) {
        const int idx = tid + i * 512;     // idx = c*512 + k
        const int c = idx >> 9;
        const int k = idx & 511;
        w2cm[c * W2_STRIDE + k] = (c < 2) ? W2[k * 2 + c] : 0.0f;
    }
    __syncthreads();

    const float* __restrict__ bCol = &w2cm[((lrow < 3) ? lrow : 3) * W2_STRIDE];

    v8f acc = {};
    for (int k0 = 0; k0 < H_MID; k0 += 4) {
        const int k = k0 + koff;
        v2f a = *(const v2f*)(W1 + m * H_MID + k);   // A[m][k], A[m][k+1]
        v2f b = *(const v2f*)(bCol + k);             // B[k][n], B[k+1][n]
        acc = __builtin_amdgcn_wmma_f32_16x16x4_f32(
                  false, a, false, b, (short)0, acc, false, false);
    }

    // C/D layout: VGPR v -> M = hi*8 + v, N = lrow.  Store column-major,
    // zero-padding columns 2..3 so the main kernel can load unconditionally.
#pragma unroll
    for (int v = 0; v < 8; ++v) {
        const int row = wave * 16 + hi * 8 + v;
        if (lrow < 4)
            ws[lrow * K_IN + row] = (lrow < 2) ? acc[v] : 0.0f;
    }

    // b_eff (512-MAC scalar loop, negligible)
    if (tid < 2) {
        float s = bias2[tid];
        for (int j = 0; j < H_MID; ++j) s += bias1[j] * W2[j * 2 + tid];
        ws[4 * K_IN + tid] = s;
    }
}

// ---------------------------------------------------------------------------
// Kernel 2: per 16-edge tile, gather concat features and compute
// logits = X(16x256) @ W_eff(256x16-padded) with 64 v_wmma_f32_16x16x4_f32,
// then 2-class softmax.  W_eff lives in LDS so the hot loop is exactly
// 1 vmem gather + 1 ds load + 1 wmma per K-step; the vmem path is reserved
// for the 512MB gather stream (unique rows ~51MB -> resident in 192MB L2).
// ---------------------------------------------------------------------------
__global__ __launch_bounds__(256) void edge_softmax_kernel(
    const float* __restrict__ feat,
    const long long* __restrict__ edges,   // (2, N_EDGES) int64
    const float* __restrict__ ws,
    float* __restrict__ out)
{
    __shared__ float weff[4 * WEFF_STRIDE];   // [c][k], cols 2..3 = 0
    __shared__ float beff[2];

    const int tid  = threadIdx.x;
    const int wave = tid >> 5;
    const int lane = tid & 31;
    const int lrow = lane & 15;
    const int hi   = lane >> 4;
    const int koff = hi * 2;

    // Stage padded W_eff (ws already holds zeros in cols 2..3): 1024 entries
#pragma unroll
    for (int i = 0; i < 4; ++i) {
        const int idx = tid + i * 256;     // idx = c*256 + k
        const int c = idx >> 8;
        const int k = idx & 255;
        weff[c * WEFF_STRIDE + k] = ws[idx];
    }
    if (tid < 2) beff[tid] = ws[4 * K_IN + tid];
    __syncthreads();

    const int  tile  = blockIdx.x * 8 + wave;         // wave-uniform
    const bool valid = tile < N_TILES;
    const int  t     = valid ? tile : (N_TILES - 1);  // clamp: EXEC stays full for WMMA

    // Edge endpoints for row M = lrow (both lane groups need the same rows)
    const long long e = (long long)t * 16 + lrow;
    const int sIdx = (int)edges[e];
    const int dIdx = (int)edges[(long long)N_EDGES + e];
    const float* __restrict__ srcP = feat + (long long)sIdx * HIDDEN;
    const float* __restrict__ dstP = feat + (long long)dIdx * HIDDEN;

    // Clamped padded W_eff column for this lane (cols 2..3 zeros; dup -> LDS broadcast)
    const float* __restrict__ wCol = &weff[((lrow < 3) ? lrow : 3) * WEFF_STRIDE];

    v8f acc = {};
#pragma unroll 8
    for (int k0 = 0; k0 < HIDDEN; k0 += 4) {       // K = 0..127  (src half)
        v2f a = *(const v2f*)(srcP + k0 + koff);
        v2f b = *(const v2f*)(wCol + k0 + koff);
        acc = __builtin_amdgcn_wmma_f32_16x16x4_f32(
                  false, a, false, b, (short)0, acc, false, false);
    }
#pragma unroll 8
    for (int k0 = 0; k0 < HIDDEN; k0 += 4) {       // K = 128..255 (dst half)
        v2f a = *(const v2f*)(dstP + k0 + koff);
        v2f b = *(const v2f*)(wCol + HIDDEN + k0 + koff);
        acc = __builtin_amdgcn_wmma_f32_16x16x4_f32(
                  false, a, false, b, (short)0, acc, false, false);
    }

    // Bias + softmax.  Edge m<8 logits live in VGPR m lanes {0,1};
    // m>=8 in VGPR m-8 lanes {16,17}.  Pair-exchange via shfl_xor(1).
    const float myb = (lrow == 0) ? beff[0] : beff[1];   // meaningful for lrow < 2

#pragma unroll
    for (int v = 0; v < 8; ++v) {
        const int m = hi * 8 + v;                  // edge index within tile
        float logit = acc[v] + myb;
        float other = __shfl_xor(logit, 1, 32);    // partner class logit
        float mx = fmaxf(logit, other);
        float ea = __expf(logit - mx);
        float eb = __expf(other - mx);
        float p  = ea / (ea + eb);
        if (valid && lrow < 2)
            out[((long long)t * 16 + m) * 2 + lrow] = p;
    }
}

// ---------------------------------------------------------------------------
extern "C" void kernel_launch(void* const* d_in, const int* in_sizes, int n_in,
                              void* d_out, int out_size, void* d_ws, size_t ws_size,
                              hipStream_t stream)
{
    const float*     feat  = (const float*)d_in[0];       // (100000, 128) f32
    const long long* edges = (const long long*)d_in[1];   // (2, 500000) i64
    const float*     W1    = (const float*)d_in[2];       // (256, 512) f32
    const float*     b1    = (const float*)d_in[3];       // (512,) f32
    const float*     W2    = (const float*)d_in[4];       // (512, 2) f32
    const float*     b2    = (const float*)d_in[5];       // (2,) f32
    float* out = (float*)d_out;
    float* ws  = (float*)d_ws;                            // needs 4104 bytes

    fold_weights_kernel<<<1, 512, 0, stream>>>(W1, b1, W2, b2, ws);

    const int blocks = (N_TILES + 7) / 8;                 // 8 tiles (waves) per block
    edge_softmax_kernel<<<blocks, 256, 0, stream>>>(feat, edges, ws, out);
}